// TensorTree_Based_4183298146462
// MI455X (gfx1250) — compile-verified
//
#include <hip/hip_runtime.h>

typedef __attribute__((ext_vector_type(2))) float v2f;
typedef __attribute__((ext_vector_type(8))) float v8f;

// Repack a 10x10x10 core (core[i,k,j], row-major) into the WMMA f32 16x16x4
// A-operand layout for the GEMM  D(16x16) = coreT(16x100) x O(100x16):
//   A(M=j, K=i*10+k) = core[i,k,j],  rows j>=10 zero-padded.
// Per wmma step t (K block 4t..4t+3): lane l holds K = 4t + (l>=16 ? 2:0) + p
// for p=0,1 in two consecutive floats -> entry index (t*32 + l)*2 + p.
__device__ __forceinline__ void fillCoreA(float* dst, const float* __restrict__ src, int tid) {
  for (int idx = tid; idx < 1600; idx += 256) {
    int p = idx & 1;
    int l = (idx >> 1) & 31;
    int t = idx >> 6;
    int K = 4 * t + ((l & 16) ? 2 : 0) + p;
    int j = l & 15;
    int i = K / 10;
    int k = K % 10;
    dst[idx] = (j < 10) ? src[(i * 10 + k) * 10 + j] : 0.0f;
  }
}

// Leaf contraction: c[r] = sum_d X[b,f,d] * FM[d,r,f,u]   (D=2, R=10, F=8, U=32)
__device__ __forceinline__ void leafContract(const float* sFM, const float* __restrict__ X,
                                             int b, int f, int u, float c[10]) {
  float x0 = X[(b * 8 + f) * 2 + 0];
  float x1 = X[(b * 8 + f) * 2 + 1];
#pragma unroll
  for (int r = 0; r < 10; ++r) {
    c[r] = x0 * sFM[((r) * 8 + f) * 32 + u] + x1 * sFM[((10 + r) * 8 + f) * 32 + u];
  }
}

// pair(a,b,core)[j] = sum_{i,k} a[i]*b[k]*core[i,k,j] for 16 rows per wave.
// Lanes n and n+16 both carry row n's full vectors. Output q is re-merged so
// every lane again holds the full 10-vector of its row.
__device__ __forceinline__ void pairContract(const float* sCoreC, const float a[10],
                                             const float bv[10], int lane, float q[10]) {
  v8f acc = {};
  bool hi = (lane & 16) != 0;
#pragma unroll
  for (int t = 0; t < 25; ++t) {
    // A operand (constant core, pre-permuted): ds_read_b64
    v2f A = *(const v2f*)(sCoreC + (t * 32 + lane) * 2);
    // B operand: per-row outer products, branch-free lane-half select so EXEC
    // stays all-1s around the WMMA. All indices are compile-time constants.
    float blo0 = a[(4 * t    ) / 10] * bv[(4 * t    ) % 10];
    float blo1 = a[(4 * t + 1) / 10] * bv[(4 * t + 1) % 10];
    float bhi0 = a[(4 * t + 2) / 10] * bv[(4 * t + 2) % 10];
    float bhi1 = a[(4 * t + 3) / 10] * bv[(4 * t + 3) % 10];
    v2f Bv;
    Bv.x = hi ? bhi0 : blo0;
    Bv.y = hi ? bhi1 : blo1;
    acc = __builtin_amdgcn_wmma_f32_16x16x4_f32(false, A, false, Bv, (short)0, acc,
                                                false, false);
  }
  // D layout: row n's out[j]: j=0..7 -> VGPR j lane n ; j=8..9 -> VGPR j-8 lane n+16.
  float m[8];
#pragma unroll
  for (int v = 0; v < 8; ++v) m[v] = __shfl_xor(acc[v], 16, 32);
#pragma unroll
  for (int v = 0; v < 8; ++v) q[v] = hi ? m[v] : acc[v];
  q[8] = hi ? acc[0] : m[0];
  q[9] = hi ? acc[1] : m[1];
}

__global__ __launch_bounds__(256) void tensortree_wmma_kernel(
    const float* __restrict__ X, const float* __restrict__ c11,
    const float* __restrict__ c12, const float* __restrict__ c13,
    const float* __restrict__ c14, const float* __restrict__ c21,
    const float* __restrict__ c22, const float* __restrict__ fm,
    const float* __restrict__ T, float* __restrict__ out) {
  __shared__ float sCore[6 * 1600];  // cores in WMMA A layout: 38.4 KB
  __shared__ float sFM[5120];        // factor_matrices [d][r][f][u]: 20 KB

  int tid = threadIdx.x;
  fillCoreA(sCore + 0,    c11, tid);
  fillCoreA(sCore + 1600, c12, tid);
  fillCoreA(sCore + 3200, c13, tid);
  fillCoreA(sCore + 4800, c14, tid);
  fillCoreA(sCore + 6400, c21, tid);
  fillCoreA(sCore + 8000, c22, tid);
  for (int idx = tid; idx < 5120; idx += 256) sFM[idx] = fm[idx];
  __syncthreads();

  int lane = tid & 31;
  int wave = blockIdx.x * 8 + (tid >> 5);   // 16 rows (one b, half the units) per wave
  int b = __builtin_amdgcn_readfirstlane(wave >> 1);
  int u = ((wave & 1) << 4) | (lane & 15);

  float cA[10], cB[10], pA[10], pB[10], q1[10], q2[10];

  leafContract(sFM, X, b, 0, u, cA);
  leafContract(sFM, X, b, 1, u, cB);
  pairContract(sCore + 0, cA, cB, lane, pA);        // p1 = pair(c0,c1,core11)
  leafContract(sFM, X, b, 2, u, cA);
  leafContract(sFM, X, b, 3, u, cB);
  pairContract(sCore + 1600, cA, cB, lane, pB);     // p2 = pair(c2,c3,core12)
  pairContract(sCore + 6400, pA, pB, lane, q1);     // q1 = pair(p1,p2,core21)

  leafContract(sFM, X, b, 4, u, cA);
  leafContract(sFM, X, b, 5, u, cB);
  pairContract(sCore + 3200, cA, cB, lane, pA);     // p3 = pair(c4,c5,core13)
  leafContract(sFM, X, b, 6, u, cA);
  leafContract(sFM, X, b, 7, u, cB);
  pairContract(sCore + 4800, cA, cB, lane, pB);     // p4 = pair(c6,c7,core14)
  pairContract(sCore + 8000, pA, pB, lane, q2);     // q2 = pair(p3,p4,core22)

  // out[b,u] = q1 . (T @ q2)   — T reads are wave-uniform -> scalar loads
  float r = 0.0f;
#pragma unroll
  for (int i = 0; i < 10; ++i) {
    float s = 0.0f;
#pragma unroll
    for (int j = 0; j < 10; ++j) s += T[i * 10 + j] * q2[j];
    r += q1[i] * s;
  }
  if (lane < 16) out[b * 32 + u] = r;  // 64B coalesced per wave
}

extern "C" void kernel_launch(void* const* d_in, const int* in_sizes, int n_in,
                              void* d_out, int out_size, void* d_ws, size_t ws_size,
                              hipStream_t stream) {
  const float* X   = (const float*)d_in[0];
  const float* c11 = (const float*)d_in[1];
  const float* c12 = (const float*)d_in[2];
  const float* c13 = (const float*)d_in[3];
  const float* c14 = (const float*)d_in[4];
  const float* c21 = (const float*)d_in[5];
  const float* c22 = (const float*)d_in[6];
  const float* fm  = (const float*)d_in[7];
  const float* T   = (const float*)d_in[8];
  float* out = (float*)d_out;

  int B = in_sizes[0] / 16;          // X is [B, 8, 2]
  int blocks = (B * 32) / 128;       // 128 rows (b,u) per 256-thread block
  tensortree_wmma_kernel<<<dim3(blocks), dim3(256), 0, stream>>>(
      X, c11, c12, c13, c14, c21, c22, fm, T, out);
}